// MLSADigitalFilterCoefficientsToMelCepstrum_17944373362778
// MI455X (gfx1250) — compile-verified
//
#include <hip/hip_runtime.h>

// out[..., m] = b[..., m] + ALPHA * b[..., m+1], last coeff unchanged.
// Implemented as a bidiagonal GEMM on the 40-wide last axis using
// V_WMMA_F32_16X16X4_F32 (exact f32 MACs, matches reference semantics).

typedef __attribute__((ext_vector_type(2))) float v2f;   // A / B operands (16x4 / 4x16 f32 = 2 VGPRs)
typedef __attribute__((ext_vector_type(8))) float v8f;   // C / D accumulator (16x16 f32 = 8 VGPRs)

#define NC 40
#define ALPHA_F 0.42f

__global__ void __launch_bounds__(256)
mlsa_wmma_kernel(const float* __restrict__ bin, float* __restrict__ out, long long nBlocks) {
    const int lane = threadIdx.x & 31;
    const int hi   = lane >> 4;      // half-wave select (K/M split per ISA layouts)
    const int nloc = lane & 15;      // row index for A, column index for B/D

    const int wavesPerWG   = blockDim.x >> 5;
    const long long wave   = (long long)blockIdx.x * wavesPerWG + (threadIdx.x >> 5);
    const long long nWaves = (long long)gridDim.x * wavesPerWG;

    // Constant bidiagonal B tiles. Tile with diagonal offset d = k0 - n0 holds
    // M[k, n] at (k = d + 2*hi + j, n = nloc):  1 if k==n, ALPHA if k==n+1, else 0.
    // B layout (4x16 f32, 2 VGPRs): VGPR0 -> K=0 (lanes 0-15) / K=2 (lanes 16-31),
    //                               VGPR1 -> K=1 / K=3; N striped across lanes.
    v2f Bm[5];
#pragma unroll
    for (int dd = 0; dd < 5; ++dd) {
#pragma unroll
        for (int j = 0; j < 2; ++j) {
            const int rel = 4 * dd + 2 * hi + j - nloc;   // k - n
            Bm[dd][j] = (rel == 0) ? 1.0f : ((rel == 1) ? ALPHA_F : 0.0f);
        }
    }
    // Column-masked variants for the last (8-wide) N-tile: zero columns n>=40.
    v2f Bt[2];
#pragma unroll
    for (int dd = 0; dd < 2; ++dd) {
        Bt[dd][0] = (nloc < 8) ? Bm[dd][0] : 0.0f;
        Bt[dd][1] = (nloc < 8) ? Bm[dd][1] : 0.0f;
    }

    for (long long blk = wave; blk < nBlocks; blk += nWaves) {
        const long long r0 = blk * 16;

        // A layout (16x4 f32, 2 VGPRs): lane L holds row M=L&15, K = 2*hi + {0,1}.
        // -> one aligned b64 load per K-block of 4; 10 blocks cover K=0..39.
        const float* rowp = bin + (r0 + (long long)nloc) * (long long)NC + 2 * hi;
        v2f A[10];
#pragma unroll
        for (int t = 0; t < 10; ++t)
            A[t] = *(const v2f*)(rowp + 4 * t);

        v8f D0 = {};  // columns  0..15
        v8f D1 = {};  // columns 16..31
        v8f D2 = {};  // columns 32..39 (upper 8 lanes of tile unused)

#pragma unroll
        for (int t = 0; t < 5; ++t)   // k0 = 4t,      d = k0 - 0  = 4t
            D0 = __builtin_amdgcn_wmma_f32_16x16x4_f32(false, A[t],     false, Bm[t], (short)0, D0, false, false);
#pragma unroll
        for (int s = 0; s < 5; ++s)   // k0 = 16 + 4s, d = k0 - 16 = 4s
            D1 = __builtin_amdgcn_wmma_f32_16x16x4_f32(false, A[4 + s], false, Bm[s], (short)0, D1, false, false);
        // k0 = 32, 36; d = 0, 4 (masked to 8 valid columns)
        D2 = __builtin_amdgcn_wmma_f32_16x16x4_f32(false, A[8], false, Bt[0], (short)0, D2, false, false);
        D2 = __builtin_amdgcn_wmma_f32_16x16x4_f32(false, A[9], false, Bt[1], (short)0, D2, false, false);

        // D layout: VGPR v -> (row v + 8*hi, col n0 + nloc); 64B half-wave-contiguous stores.
        float* op = out + r0 * (long long)NC;
#pragma unroll
        for (int v = 0; v < 8; ++v) {
            const long long roff = (long long)(v + 8 * hi) * NC;
            op[roff + nloc]      = D0[v];
            op[roff + 16 + nloc] = D1[v];
            if (nloc < 8)
                op[roff + 32 + nloc] = D2[v];
        }
    }
}

// Generic elementwise tail for rows not covered by 16-row blocks (empty for the
// reference problem size, kept for generality).
__global__ void mlsa_tail_kernel(const float* __restrict__ bin, float* __restrict__ out,
                                 long long start, long long total) {
    long long i = start + (long long)blockIdx.x * blockDim.x + threadIdx.x;
    if (i >= total) return;
    const long long m = i % NC;
    const float v  = bin[i];
    const float nx = (m < NC - 1 && i + 1 < total) ? bin[i + 1] : 0.0f;
    out[i] = fmaf(ALPHA_F, nx, v);
}

extern "C" void kernel_launch(void* const* d_in, const int* in_sizes, int n_in,
                              void* d_out, int out_size, void* d_ws, size_t ws_size,
                              hipStream_t stream) {
    const float* bin = (const float*)d_in[0];
    float* out = (float*)d_out;

    const long long total   = (long long)in_sizes[0];
    const long long rows    = total / NC;
    const long long nBlocks = rows / 16;

    if (nBlocks > 0) {
        const int threads    = 256;                    // 8 waves (wave32)
        const int wavesPerWG = threads / 32;
        long long wgNeeded   = (nBlocks + wavesPerWG - 1) / wavesPerWG;
        int grid = (int)(wgNeeded < 4096 ? wgNeeded : 4096);  // persistent waves, grid-stride
        mlsa_wmma_kernel<<<grid, threads, 0, stream>>>(bin, out, nBlocks);
    }

    const long long done = nBlocks * 16 * (long long)NC;
    if (done < total) {
        const long long tail = total - done;
        const int threads = 256;
        const long long grid = (tail + threads - 1) / threads;
        mlsa_tail_kernel<<<(int)grid, threads, 0, stream>>>(bin, out, done, total);
    }
}